// SPDNetwork_6717328851873
// MI455X (gfx1250) — compile-verified
//
#include <hip/hip_runtime.h>

// SPDNet forward, algebraically fused for MI455X (gfx1250, wave32, WMMA).
//
// rec_mat is the identity here (lambda_min >= 1e-2 > eps = 1e-4 by Cauchy
// interlacing with orthonormal projections), so the network reduces to
//   X3 = logm(W123^T x W123), W123 = w1@w2@w3 (400x50); logits = vec(X3)@fc_w.
// The big GEMM streams x (164 MB) once -> HBM-bound (~10 us at 23.3 TB/s);
// W123 (100 KB) stays hot in the 192 MB L2. Matrix ops use
// V_WMMA_F32_16X16X4_F32 (native f32). logm uses a per-matrix cyclic Jacobi
// eigensolver in LDS with one wave32 per 50x50 matrix.

typedef float v2f __attribute__((ext_vector_type(2)));
typedef float v8f __attribute__((ext_vector_type(8)));

#define BATCH 256
#define N0    400   // input dim
#define N3    50    // final dim
#define NP    64    // 50 padded to 64 for 16x16 WMMA tiling

// workspace layout (in floats)
static constexpr size_t OFF_W12  = 0;                         // [400,100]
static constexpr size_t OFF_W123 = OFF_W12  + 400u * 100u;    // [400,64] zero-padded
static constexpr size_t OFF_T    = OFF_W123 + 400u * 64u;     // [256,400,64]
static constexpr size_t OFF_Y    = OFF_T    + 256u * 400u * 64u; // [256,64,64]
static constexpr size_t OFF_X3   = OFF_Y    + 256u * 64u * 64u;  // [256,50,50]

// ---------------- weight fusion (tiny) ----------------
__global__ void k_w12(const float* __restrict__ w1, const float* __restrict__ w2,
                      float* __restrict__ W12) {
  int idx = blockIdx.x * blockDim.x + threadIdx.x;
  if (idx >= 400 * 100) return;
  int i = idx / 100, j = idx % 100;
  float acc = 0.f;
  for (int k = 0; k < 200; ++k) acc = fmaf(w1[i * 200 + k], w2[k * 100 + j], acc);
  W12[idx] = acc;
}

__global__ void k_w123(const float* __restrict__ W12, const float* __restrict__ w3,
                       float* __restrict__ W123) {
  int idx = blockIdx.x * blockDim.x + threadIdx.x;
  if (idx >= 400 * NP) return;
  int i = idx / NP, j = idx % NP;
  float acc = 0.f;
  if (j < N3)
    for (int k = 0; k < 100; ++k) acc = fmaf(W12[i * 100 + k], w3[k * N3 + j], acc);
  W123[idx] = acc;  // cols 50..63 zero-padded
}

// ---------------- T_b = x_b @ W123  (WMMA f32 16x16x4) ----------------
// grid (25 row-tiles, 256 batches), block 128 = 4 waves; wave w handles n-tile w.
// Per ISA 7.12.2: 32-bit A 16x4: lanes 0-15 hold K={0,1}, lanes 16-31 K={2,3};
// B 4x16 mirrored; C/D 16x16: VGPR r -> row (r + 8*half), col = lane%16.
__global__ void k_xw(const float* __restrict__ x, const float* __restrict__ W,
                     float* __restrict__ T) {
  int b    = blockIdx.y;
  int m0   = blockIdx.x * 16;
  int wave = threadIdx.x >> 5;
  int lane = threadIdx.x & 31;
  int n0   = wave * 16;
  int half = lane >> 4;
  int l15  = lane & 15;
  const float* xb = x + (size_t)b * N0 * N0;

  v8f c = {};
  for (int kk = 0; kk < N0; kk += 4) {
    int kb = kk + half * 2;
    v2f a, bb;
    a.x  = xb[(size_t)(m0 + l15) * N0 + kb];
    a.y  = xb[(size_t)(m0 + l15) * N0 + kb + 1];
    bb.x = W[(size_t)kb * NP + n0 + l15];
    bb.y = W[(size_t)(kb + 1) * NP + n0 + l15];
    c = __builtin_amdgcn_wmma_f32_16x16x4_f32(false, a, false, bb, (short)0, c,
                                              false, false);
  }
  float* Tb = T + (size_t)b * N0 * NP;
#pragma unroll
  for (int r = 0; r < 8; ++r)
    Tb[(size_t)(m0 + r + half * 8) * NP + n0 + l15] = c[r];
}

// ---------------- Y_b = W123^T @ T_b  (WMMA f32 16x16x4) ----------------
// grid (4 n-tiles, 256 batches), block 128; wave = m-tile. A[m][k] = W[k][m].
__global__ void k_wtt(const float* __restrict__ W, const float* __restrict__ T,
                      float* __restrict__ Y) {
  int b    = blockIdx.y;
  int n0   = blockIdx.x * 16;
  int wave = threadIdx.x >> 5;
  int lane = threadIdx.x & 31;
  int m0   = wave * 16;
  int half = lane >> 4;
  int l15  = lane & 15;
  const float* Tb = T + (size_t)b * N0 * NP;

  v8f c = {};
  for (int kk = 0; kk < N0; kk += 4) {
    int kb = kk + half * 2;
    v2f a, bb;
    a.x  = W[(size_t)kb * NP + m0 + l15];        // transposed read
    a.y  = W[(size_t)(kb + 1) * NP + m0 + l15];
    bb.x = Tb[(size_t)kb * NP + n0 + l15];
    bb.y = Tb[(size_t)(kb + 1) * NP + n0 + l15];
    c = __builtin_amdgcn_wmma_f32_16x16x4_f32(false, a, false, bb, (short)0, c,
                                              false, false);
  }
  float* Yb = Y + (size_t)b * NP * NP;
#pragma unroll
  for (int r = 0; r < 8; ++r)
    Yb[(size_t)(m0 + r + half * 8) * NP + n0 + l15] = c[r];
}

// ---------------- logm via cyclic Jacobi, one wave32 per matrix ----------------
// Single-wave workgroup: barrier ops lower to S_NOP (ISA 3.1) but keep the
// compiler from reordering LDS traffic across the cross-lane dependencies.
__global__ void k_logm(const float* __restrict__ Y, float* __restrict__ X3) {
  __shared__ float A[N3][N3 + 2];
  __shared__ float V[N3][N3 + 2];
  __shared__ float logd[N3];
  int b = blockIdx.x;
  int t = threadIdx.x;  // 0..31
  const float* Yb = Y + (size_t)b * NP * NP;

  for (int idx = t; idx < N3 * N3; idx += 32) {
    int i = idx / N3, j = idx % N3;
    A[i][j] = Yb[(size_t)i * NP + j];
    V[i][j] = (i == j) ? 1.f : 0.f;
  }
  __syncthreads();

  for (int sweep = 0; sweep < 10; ++sweep) {
    for (int p = 0; p < N3 - 1; ++p) {
      for (int q = p + 1; q < N3; ++q) {
        // all lanes read the same LDS cells -> uniform c,s (broadcast)
        float apq = A[p][q], app = A[p][p], aqq = A[q][q];
        float c = 1.f, s = 0.f;
        if (fabsf(apq) > 1e-13f * (fabsf(app) + fabsf(aqq)) + 1e-30f) {
          float theta = 0.5f * (aqq - app) / apq;
          float tt = copysignf(1.f, theta) /
                     (fabsf(theta) + sqrtf(fmaf(theta, theta, 1.f)));
          c = rsqrtf(fmaf(tt, tt, 1.f));
          s = tt * c;
        }
        __syncthreads();
        // A <- A*J, V <- V*J  (column rotation)
        for (int k = t; k < N3; k += 32) {
          float akp = A[k][p], akq = A[k][q];
          A[k][p] = fmaf(c, akp, -s * akq);
          A[k][q] = fmaf(s, akp,  c * akq);
          float vkp = V[k][p], vkq = V[k][q];
          V[k][p] = fmaf(c, vkp, -s * vkq);
          V[k][q] = fmaf(s, vkp,  c * vkq);
        }
        __syncthreads();
        // A <- J^T * A  (row rotation)
        for (int k = t; k < N3; k += 32) {
          float apk = A[p][k], aqk = A[q][k];
          A[p][k] = fmaf(c, apk, -s * aqk);
          A[q][k] = fmaf(s, apk,  c * aqk);
        }
        __syncthreads();
      }
    }
  }

  for (int k = t; k < N3; k += 32) logd[k] = logf(fmaxf(A[k][k], 1e-12f));
  __syncthreads();

  // X3 = V diag(log lambda) V^T
  float* o = X3 + (size_t)b * N3 * N3;
  for (int idx = t; idx < N3 * N3; idx += 32) {
    int i = idx / N3, j = idx % N3;
    float acc = 0.f;
    for (int k = 0; k < N3; ++k) acc = fmaf(V[i][k] * logd[k], V[j][k], acc);
    o[idx] = acc;
  }
}

// ---------------- logits = vec(X3) @ fc_w ----------------
__global__ void k_fc(const float* __restrict__ X3, const float* __restrict__ fcw,
                     float* __restrict__ out) {
  __shared__ float red[256 * 7];
  int b = blockIdx.x, t = threadIdx.x;
  const float* f = X3 + (size_t)b * 2500;
  float acc[7] = {0.f, 0.f, 0.f, 0.f, 0.f, 0.f, 0.f};
  for (int i = t; i < 2500; i += 256) {
    float v = f[i];
#pragma unroll
    for (int j = 0; j < 7; ++j) acc[j] = fmaf(v, fcw[i * 7 + j], acc[j]);
  }
#pragma unroll
  for (int j = 0; j < 7; ++j) red[t * 7 + j] = acc[j];
  __syncthreads();
  for (int s2 = 128; s2 > 0; s2 >>= 1) {
    if (t < s2) {
#pragma unroll
      for (int j = 0; j < 7; ++j) red[t * 7 + j] += red[(t + s2) * 7 + j];
    }
    __syncthreads();
  }
  if (t < 7) out[b * 7 + t] = red[t];
}

extern "C" void kernel_launch(void* const* d_in, const int* in_sizes, int n_in,
                              void* d_out, int out_size, void* d_ws, size_t ws_size,
                              hipStream_t stream) {
  const float* x   = (const float*)d_in[0];  // [256,400,400]
  const float* w1  = (const float*)d_in[1];  // [400,200]
  const float* w2  = (const float*)d_in[2];  // [200,100]
  const float* w3  = (const float*)d_in[3];  // [100,50]
  const float* fcw = (const float*)d_in[4];  // [2500,7]
  float* out = (float*)d_out;                // [256,7]

  float* ws   = (float*)d_ws;
  float* W12  = ws + OFF_W12;
  float* W123 = ws + OFF_W123;
  float* T    = ws + OFF_T;
  float* Y    = ws + OFF_Y;
  float* X3   = ws + OFF_X3;

  k_w12 <<<(400 * 100 + 255) / 256, 256, 0, stream>>>(w1, w2, W12);
  k_w123<<<(400 * NP  + 255) / 256, 256, 0, stream>>>(W12, w3, W123);
  k_xw  <<<dim3(N0 / 16, BATCH), 128, 0, stream>>>(x, W123, T);
  k_wtt <<<dim3(NP / 16, BATCH), 128, 0, stream>>>(W123, T, Y);
  k_logm<<<BATCH, 32, 0, stream>>>(Y, X3);
  k_fc  <<<BATCH, 256, 0, stream>>>(X3, fcw, out);
}